// GCNLayer_27986006901490
// MI455X (gfx1250) — compile-verified
//
#include <hip/hip_runtime.h>

typedef float v2f __attribute__((ext_vector_type(2)));
typedef float v8f __attribute__((ext_vector_type(8)));

#define GCN_N_NODES 50000
#define GCN_F 32

// ---------------------------------------------------------------------------
// Kernel 1: zero the workspace accumulators (node_sum, node_h, deg).
// Regular (RT) stores: this data is immediately re-used, keep it in L2.
// ---------------------------------------------------------------------------
__global__ void __launch_bounds__(256) gcn_zero_f32(float* __restrict__ p, long n) {
  long i = (long)blockIdx.x * blockDim.x + threadIdx.x;
  if (i < n) p[i] = 0.0f;
}

// ---------------------------------------------------------------------------
// Kernel 2: node_sum[dst[e]][f] += inputs[e][f];  deg[dst[e]] += 1
// One lane per (edge, feature): input reads fully coalesced and NON-TEMPORAL
// (205 MB streamed once -> don't evict the 6.4 MB node accumulators from L2).
// Atomics land in L2 (node arrays << 192 MB L2).
// ---------------------------------------------------------------------------
__global__ void __launch_bounds__(256) gcn_scatter_sum_deg(
    const float* __restrict__ inputs, const int* __restrict__ dst,
    float* __restrict__ node_sum, float* __restrict__ deg, long E) {
  long t = (long)blockIdx.x * blockDim.x + threadIdx.x;
  long e = t >> 5;
  int  f = (int)(t & 31);
  if (e >= E) return;
  int   d = dst[e];
  float x = __builtin_nontemporal_load(inputs + e * GCN_F + f);
  atomicAdd(node_sum + (long)d * GCN_F + f, x);
  if (f == 0) atomicAdd(deg + d, 1.0f);
}

// ---------------------------------------------------------------------------
// Kernel 3: node_mean = node_sum / max(deg, 1)   (in place)
// ---------------------------------------------------------------------------
__global__ void __launch_bounds__(256) gcn_finalize_mean(
    float* __restrict__ node_sum, const float* __restrict__ deg, int n_nodes) {
  long t = (long)blockIdx.x * blockDim.x + threadIdx.x;
  long n = t >> 5;
  int  f = (int)(t & 31);
  if (n >= n_nodes) return;
  float dv = deg[n];
  dv = dv < 1.0f ? 1.0f : dv;
  node_sum[n * GCN_F + f] /= dv;
}

// ---------------------------------------------------------------------------
// Kernel 4: node_h[dst[e]][f] += node_mean[src[e]][f]
// Gather + atomic scatter, both L2-resident.
// ---------------------------------------------------------------------------
__global__ void __launch_bounds__(256) gcn_scatter_node_h(
    const float* __restrict__ node_mean, const int* __restrict__ src,
    const int* __restrict__ dst, float* __restrict__ node_h, long E) {
  long t = (long)blockIdx.x * blockDim.x + threadIdx.x;
  long e = t >> 5;
  int  f = (int)(t & 31);
  if (e >= E) return;
  int s = src[e];
  int d = dst[e];
  atomicAdd(node_h + (long)d * GCN_F + f, node_mean[(long)s * GCN_F + f]);
}

// ---------------------------------------------------------------------------
// Kernel 5: out[e] = (0.5*(node_h[src[e]] + node_h[dst[e]])) @ W^T + b
// One wave (32 lanes) computes a 16-edge x 32-feature output tile using
// V_WMMA_F32_16X16X4_F32, K = 32 -> 8 chained WMMAs per N-tile, 2 N-tiles.
// Bias is folded into the initial accumulator (D column N is constant per
// lane across all 8 C/D VGPRs). Output stores are non-temporal (streamed).
//
// A-matrix (16x4, M x K) ISA layout per K-step s (K base = 4*s):
//   lanes 0-15  (row M = lane)    : VGPR0 = K+0, VGPR1 = K+1
//   lanes 16-31 (row M = lane-16) : VGPR0 = K+2, VGPR1 = K+3
// B-matrix (4x16, K x N), B[k][n] = W[n][k] (out = A @ W^T):
//   lanes 0-15  (col N = lane)    : VGPR0 = K+0, VGPR1 = K+1
//   lanes 16-31 (col N = lane-16) : VGPR0 = K+2, VGPR1 = K+3
// C/D (16x16 f32): VGPR v, lanes 0-15 -> (M=v,   N=lane),
//                          lanes 16-31 -> (M=v+8, N=lane-16)
// ---------------------------------------------------------------------------
__global__ void __launch_bounds__(256) gcn_edge_proj_wmma(
    const float* __restrict__ node_h, const int* __restrict__ src,
    const int* __restrict__ dst, const float* __restrict__ W,
    const float* __restrict__ bias, float* __restrict__ out, long E) {
  const int  lane = threadIdx.x & 31;
  const int  wave = threadIdx.x >> 5;
  const long tile = (long)blockIdx.x * (blockDim.x >> 5) + wave;
  const long base = tile * 16;
  if (base >= E) return;  // uniform per wave: EXEC stays all-1s for WMMA

  const int half = lane >> 4;  // 0 -> K+0/K+1 ; 1 -> K+2/K+3
  const int r    = lane & 15;  // A row within tile / B,C,D column

  long e = base + r;
  if (e >= E) e = E - 1;  // clamp loads on the tail tile; stores are guarded
  const float* hs = node_h + (long)src[e] * GCN_F;
  const float* hd = node_h + (long)dst[e] * GCN_F;
  const float* w0 = W + (long)r * GCN_F;         // W row n = r       (N-tile 0)
  const float* w1 = W + (long)(r + 16) * GCN_F;  // W row n = r + 16  (N-tile 1)

  // Seed accumulators with the bias (column N = r / r+16 for this lane).
  const float bv0 = bias[r];
  const float bv1 = bias[r + 16];
  v8f acc0, acc1;
#pragma unroll
  for (int v = 0; v < 8; ++v) { acc0[v] = bv0; acc1[v] = bv1; }

#pragma unroll
  for (int s = 0; s < 8; ++s) {
    const int kc = 4 * s + 2 * half;
    v2f as = *(const v2f*)(hs + kc);
    v2f ad = *(const v2f*)(hd + kc);
    v2f a  = 0.5f * (as + ad);            // edge_h on the fly
    v2f b0 = *(const v2f*)(w0 + kc);      // B operand, N-tile 0
    v2f b1 = *(const v2f*)(w1 + kc);      // B operand, N-tile 1
    acc0 = __builtin_amdgcn_wmma_f32_16x16x4_f32(
        false, a, false, b0, (short)0, acc0, false, false);
    acc1 = __builtin_amdgcn_wmma_f32_16x16x4_f32(
        false, a, false, b1, (short)0, acc1, false, false);
  }

#pragma unroll
  for (int v = 0; v < 8; ++v) {
    const long m = base + v + 8 * half;
    if (m < E) {
      __builtin_nontemporal_store(acc0[v], out + m * GCN_F + r);
      __builtin_nontemporal_store(acc1[v], out + m * GCN_F + r + 16);
    }
  }
}

// ---------------------------------------------------------------------------
// Launcher
// ---------------------------------------------------------------------------
extern "C" void kernel_launch(void* const* d_in, const int* in_sizes, int n_in,
                              void* d_out, int out_size, void* d_ws, size_t ws_size,
                              hipStream_t stream) {
  const float* inputs = (const float*)d_in[0];
  const int*   src    = (const int*)d_in[1];
  const int*   dst    = (const int*)d_in[2];
  const float* W      = (const float*)d_in[3];
  const float* bias   = (const float*)d_in[4];
  float*       out    = (float*)d_out;
  const long   E      = (long)in_sizes[1];

  // Workspace layout (floats): node_sum[N*F] | node_h[N*F] | deg[N]  (~13 MB)
  float* node_sum = (float*)d_ws;
  float* node_h   = node_sum + (long)GCN_N_NODES * GCN_F;
  float* deg      = node_h   + (long)GCN_N_NODES * GCN_F;

  const long zn = (long)GCN_N_NODES * GCN_F * 2 + GCN_N_NODES;
  gcn_zero_f32<<<(int)((zn + 255) / 256), 256, 0, stream>>>((float*)d_ws, zn);

  const long tE = E * GCN_F;
  gcn_scatter_sum_deg<<<(int)((tE + 255) / 256), 256, 0, stream>>>(
      inputs, dst, node_sum, deg, E);

  const long tN = (long)GCN_N_NODES * GCN_F;
  gcn_finalize_mean<<<(int)((tN + 255) / 256), 256, 0, stream>>>(
      node_sum, deg, GCN_N_NODES);

  gcn_scatter_node_h<<<(int)((tE + 255) / 256), 256, 0, stream>>>(
      node_sum, src, dst, node_h, E);

  const long tiles  = (E + 15) / 16;
  const int  blocks = (int)((tiles + 7) / 8);  // 8 waves / 256-thread block
  gcn_edge_proj_wmma<<<blocks, 256, 0, stream>>>(
      node_h, src, dst, W, bias, out, E);
}